// AIST_25357486916382
// MI455X (gfx1250) — compile-verified
//
#include <hip/hip_runtime.h>
#include <math.h>

#define BSZ   16
#define NN    2048
#define FOUT  40

typedef __attribute__((ext_vector_type(2))) float v2f;
typedef __attribute__((ext_vector_type(8))) float v8f;

// One wave (32 lanes) handles one (batch b, 16-row i-tile). 4 waves/block share b
// so x[b,:] is staged once in LDS. grid = 16 b * 32 blocks, 4 waves each = 2048 tiles.
__global__ __launch_bounds__(128)
void AIST_gat_kernel(const float* __restrict__ x,
                     const float* __restrict__ adj,
                     const float* __restrict__ W,
                     const float* __restrict__ a,
                     float* __restrict__ out)
{
    __shared__ __align__(16) float xsh[NN];
    __shared__ float numbuf[4][16];
    __shared__ float red[128];

    const int tid  = threadIdx.x;
    const int wave = tid >> 5;
    const int lane = tid & 31;
    const int hf   = lane >> 4;     // lane half: selects K offset (A/B) and M offset (C/D)
    const int il   = lane & 15;

    const int b     = blockIdx.x >> 5;                     // [0,16)
    const int itile = ((blockIdx.x & 31) << 2) + wave;     // [0,128)

    // Stage x[b, :] (8KB) into LDS + per-thread partial sum for the fallback mean.
    float p = 0.f;
    for (int j = tid; j < NN; j += 128) {
        float v = x[b * NN + j];
        xsh[j] = v;
        p += v;
    }
    red[tid] = p;
    __syncthreads();
    // Deterministic block tree-reduction (no FP atomics: harness requires
    // bit-identical output on every replay).
    #pragma unroll
    for (int s = 64; s > 0; s >>= 1) {
        if (tid < s) red[tid] += red[tid + s];
        __syncthreads();
    }
    const float xtot = red[0];

    // s1 = W . a[:40], s2 = W . a[40:]  (uniform -> scalar loads)
    float s1 = 0.f, s2 = 0.f;
    for (int f = 0; f < FOUT; ++f) {
        float wf = W[f];
        s1 = fmaf(wf, a[f], s1);
        s2 = fmaf(wf, a[FOUT + f], s2);
    }

    const int   i  = (itile << 4) + il;                    // row within graph
    const float xi = xsh[i];
    const float fi = s1 * xi;
    const int koff = hf << 1;                              // A/B layout: K = 2*half + vgpr
    const float* __restrict__ adjRow = adj + (size_t)i * NN + koff;
    const float* __restrict__ xrow   = xsh + koff;

    v8f c0 = {0,0,0,0,0,0,0,0};
    v8f c1 = {0,0,0,0,0,0,0,0};
    float den = 0.f;

    // Single-pass masked softmax accumulation. Each trip covers 8 j's
    // (two 4-wide WMMA steps on independent accumulators).
    // valid => t>0 => w=exp(t)>=1, so (den>0) <=> "row has any valid edge".
    #pragma unroll 4
    for (int j0 = 0; j0 < NN; j0 += 8) {
        {
            v2f av = *(const v2f*)(adjRow + j0);
            v2f xv = *(const v2f*)(xrow + j0);
            float t0 = fmaf(s2, xv.x, fi);
            float t1 = fmaf(s2, xv.y, fi);
            bool v0 = (av.x > 0.f) && (t0 > 0.f);
            bool v1 = (av.y > 0.f) && (t1 > 0.f);
            float w0 = v0 ? __expf(t0) : 0.f;   // t bounded (|t|<~11): fast exp is safe
            float w1 = v1 ? __expf(t1) : 0.f;
            den += w0 + w1;
            v2f A = {w0, w1};
            c0 = __builtin_amdgcn_wmma_f32_16x16x4_f32(false, A, false, xv,
                                                       (short)0, c0, false, false);
        }
        {
            v2f av = *(const v2f*)(adjRow + j0 + 4);
            v2f xv = *(const v2f*)(xrow + j0 + 4);
            float t0 = fmaf(s2, xv.x, fi);
            float t1 = fmaf(s2, xv.y, fi);
            bool v0 = (av.x > 0.f) && (t0 > 0.f);
            bool v1 = (av.y > 0.f) && (t1 > 0.f);
            float w0 = v0 ? __expf(t0) : 0.f;
            float w1 = v1 ? __expf(t1) : 0.f;
            den += w0 + w1;
            v2f A = {w0, w1};
            c1 = __builtin_amdgcn_wmma_f32_16x16x4_f32(false, A, false, xv,
                                                       (short)0, c1, false, false);
        }
    }

    // Combine the two lane-halves (they cover j%4 in {0,1} vs {2,3}).
    v8f c = c0 + c1;
    den  += __shfl_xor(den, 16, 32);

    // Re-lane the WMMA numerator: D vgpr r, half h -> row M = r + 8h
    // (all 16 columns identical since B columns were identical).
    #pragma unroll
    for (int r = 0; r < 8; ++r) numbuf[wave][(hf << 3) + r] = c[r];
    asm volatile("" ::: "memory");   // keep the LDS load after the stores
    float num = numbuf[wave][il];

    // all-masked row => reference softmax is uniform => y = mean(x[b,:])
    float y = (den > 0.f) ? (num / den) : (xtot * (1.0f / (float)NN));

    // out[b,i,:] = elu(y * W) via rank-1 WMMA outer products, 3 column tiles.
    // A[i][0] = y_i (lanes 0-15, vgpr0 = K0), all other K rows zero.
    const float  yA      = hf ? 0.f : y;
    const size_t rowBase = (size_t)b * NN + (size_t)itile * 16;

    #pragma unroll
    for (int ft = 0; ft < 3; ++ft) {
        const int f = ft * 16 + il;
        const float wB = (hf == 0 && f < FOUT) ? W[f] : 0.f;   // B[0][n] = W[f]
        v2f A  = { yA, 0.f };
        v2f Bv = { wB, 0.f };
        v8f z  = {0,0,0,0,0,0,0,0};
        z = __builtin_amdgcn_wmma_f32_16x16x4_f32(false, A, false, Bv,
                                                  (short)0, z, false, false);
        if (f < FOUT) {
            #pragma unroll
            for (int r = 0; r < 8; ++r) {
                float zz = z[r];                    // = y[M] * W[f], M = r + 8*hf
                float o  = zz > 0.f ? zz : expm1f(zz);
                out[(rowBase + (size_t)((hf << 3) + r)) * FOUT + f] = o;
            }
        }
    }
}

extern "C" void kernel_launch(void* const* d_in, const int* in_sizes, int n_in,
                              void* d_out, int out_size, void* d_ws, size_t ws_size,
                              hipStream_t stream) {
    // setup_inputs order: x, adj, ext_input(unused), side_input(unused), W, a
    const float* x   = (const float*)d_in[0];
    const float* adj = (const float*)d_in[1];
    const float* W   = (const float*)d_in[4];
    const float* a   = (const float*)d_in[5];
    float* out = (float*)d_out;

    dim3 block(128);           // 4 waves (wave32)
    dim3 grid(BSZ * 32);       // 16 batches * 32 blocks * 4 waves = 2048 row-tiles
    AIST_gat_kernel<<<grid, block, 0, stream>>>(x, adj, W, a, out);
}